// LowpassFilter_13743895347209
// MI455X (gfx1250) — compile-verified
//
#include <hip/hip_runtime.h>
#include <math.h>

typedef __attribute__((ext_vector_type(2))) float v2f;
typedef __attribute__((ext_vector_type(4))) float v4f;
typedef __attribute__((ext_vector_type(8))) float v8f;

#ifndef M_PI
#define M_PI 3.14159265358979323846
#endif

#define SEG_LEN 1024   // outputs per segment (parallel across blocks)
#define WARMUP  64     // overlap-discard warm-up; 0.42^64 ~ 1e-24 << f32 eps

__device__ __forceinline__ float lane_gather(float v, int idx) {
  // gather g-value held by lane `idx` (idx in [0,16], uniform-structure bpermute)
  return __shfl(v, idx < 0 ? 0 : idx, 32);
}

__global__ __launch_bounds__(256) void biquad_wmma_kernel(
    const float* __restrict__ x, float* __restrict__ y, int T)
{
  const int lane  = threadIdx.x & 31;
  const int wave  = threadIdx.x >> 5;   // 8 waves = 8 batch groups of 16
  const int n     = lane & 15;          // batch column within group; also A's M row
  const int hi    = lane >> 4;          // half-wave selector in WMMA layouts
  const int batch = (wave << 4) + n;    // this lane's batch row (B = 128)

  const float* __restrict__ xr = x + (size_t)batch * T;
  float*       __restrict__ yr = y + (size_t)batch * T;

  // --- biquad coefficients (double, rounded to f32 like the reference) ---
  const double w0    = 2.0 * M_PI * 10000.0 / 44100.0;
  const double alpha = sin(w0) / (2.0 * 0.707);
  const double cw    = cos(w0);
  const double a0d   = 1.0 + alpha;
  const float b0 = (float)(((1.0 - cw) * 0.5) / a0d);
  const float b1 = (float)((1.0 - cw) / a0d);
  const float b2 = b0;
  const float a1 = (float)((-2.0 * cw) / a0d);
  const float a2 = (float)((1.0 - alpha) / a0d);

  // --- g[k]: AR impulse response of 1/(1 + a1 z^-1 + a2 z^-2); lane i holds g[i], i<=16
  float gd = 0.0f;
  {
    float gm1 = 0.0f;   // g[k-1]
    float cur = 1.0f;   // g[0]
    #pragma unroll
    for (int k = 0; k <= 16; ++k) {
      if (k == lane) gd = cur;
      float nxt = -a1 * cur - a2 * gm1;
      gm1 = cur; cur = nxt;
    }
  }

  // --- A-matrix fragments: L[M][k] = (k<=M) ? g[M-k] : 0, per the 32-bit 16x4
  // A layout (lanes 0-15: K = {0,1} in v.xy; lanes 16-31: K = {2,3}).
  v2f A0, A1, A2, A3;
  {
    int k0;
    k0 =  0 + 2*hi;
    A0.x = (n-k0   >= 0) ? lane_gather(gd, n-k0  ) : 0.0f;
    A0.y = (n-k0-1 >= 0) ? lane_gather(gd, n-k0-1) : 0.0f;
    k0 =  4 + 2*hi;
    A1.x = (n-k0   >= 0) ? lane_gather(gd, n-k0  ) : 0.0f;
    A1.y = (n-k0-1 >= 0) ? lane_gather(gd, n-k0-1) : 0.0f;
    k0 =  8 + 2*hi;
    A2.x = (n-k0   >= 0) ? lane_gather(gd, n-k0  ) : 0.0f;
    A2.y = (n-k0-1 >= 0) ? lane_gather(gd, n-k0-1) : 0.0f;
    k0 = 12 + 2*hi;
    A3.x = (n-k0   >= 0) ? lane_gather(gd, n-k0  ) : 0.0f;
    A3.y = (n-k0-1 >= 0) ? lane_gather(gd, n-k0-1) : 0.0f;
  }

  // --- state-injection coefficients per accumulator reg r (row M = 8*hi + r):
  //     y_zi[M] = g[M+1]*s1 + (-a2*g[M])*s2   with s1=y[t0-1], s2=y[t0-2]
  float c1[8], c2[8];
  #pragma unroll
  for (int r = 0; r < 8; ++r) {
    int m = 8*hi + r;
    c1[r] = lane_gather(gd, m + 1);
    c2[r] = -a2 * lane_gather(gd, m);
  }

  const int seg_start = blockIdx.x * SEG_LEN;
  const int seg_end   = min(seg_start + SEG_LEN, T);
  int t0 = max(seg_start - WARMUP, 0);

  float s1 = 0.0f, s2 = 0.0f;  // exact zero state at t=0; converges during warm-up otherwise

  for (; t0 < seg_end; t0 += 16) {
    if (t0 + 256 < T) __builtin_prefetch(xr + t0 + 256, 0, 1);  // global_prefetch_b8

    // FIR: this lane needs w at k = {klo, klo+1}, klo = 4*kb + 2*hi (B-matrix layout).
    // Each pair needs x[t0+klo-2 .. t0+klo+1]: two 8B-aligned float2 loads.
    v2f Bm0, Bm1, Bm2, Bm3;
    #pragma unroll
    for (int kb = 0; kb < 4; ++kb) {
      const int klo  = 4*kb + 2*hi;
      const int base = t0 + klo - 2;
      v2f xa;
      if (base >= 0) xa = *(const v2f*)(xr + base);
      else           { xa.x = 0.0f; xa.y = 0.0f; }           // x[t<0] == 0 (zero init state)
      const v2f xb = *(const v2f*)(xr + base + 2);
      v2f wv;
      wv.x = b0 * xb.x + b1 * xa.y + b2 * xa.x;
      wv.y = b0 * xb.y + b1 * xb.x + b2 * xa.y;
      if (kb == 0) Bm0 = wv; else if (kb == 1) Bm1 = wv;
      else if (kb == 2) Bm2 = wv; else Bm3 = wv;
    }

    // Seed C with the zero-input (state) response, then K=16 via 4 chained K=4 WMMAs.
    v8f c;
    #pragma unroll
    for (int r = 0; r < 8; ++r) c[r] = c1[r] * s1 + c2[r] * s2;

    c = __builtin_amdgcn_wmma_f32_16x16x4_f32(false, A0, false, Bm0, (short)0, c, false, false);
    c = __builtin_amdgcn_wmma_f32_16x16x4_f32(false, A1, false, Bm1, (short)0, c, false, false);
    c = __builtin_amdgcn_wmma_f32_16x16x4_f32(false, A2, false, Bm2, (short)0, c, false, false);
    c = __builtin_amdgcn_wmma_f32_16x16x4_f32(false, A3, false, Bm3, (short)0, c, false, false);

    if (t0 >= seg_start) {  // drop warm-up chunks
      v4f lo; lo.x = c[0]; lo.y = c[1]; lo.z = c[2]; lo.w = c[3];
      v4f hv; hv.x = c[4]; hv.y = c[5]; hv.z = c[6]; hv.w = c[7];
      *(v4f*)(yr + t0 + 8*hi)     = lo;   // rows M = 8*hi + 0..3 -> 4 consecutive t
      *(v4f*)(yr + t0 + 8*hi + 4) = hv;
    }

    // carry state: y[15,:] lives in c[7] lanes 16-31, y[14,:] in c[6] lanes 16-31
    s1 = __shfl(c[7], 16 + n, 32);
    s2 = __shfl(c[6], 16 + n, 32);
  }
}

extern "C" void kernel_launch(void* const* d_in, const int* in_sizes, int n_in,
                              void* d_out, int out_size, void* d_ws, size_t ws_size,
                              hipStream_t stream) {
  (void)n_in; (void)out_size; (void)d_ws; (void)ws_size;
  const float* clip = (const float*)d_in[0];
  float* out = (float*)d_out;
  const int B = 128;                       // fixed by reference
  const int T = in_sizes[0] / B;           // 160000
  const int nseg = (T + SEG_LEN - 1) / SEG_LEN;
  // 256 threads = 8 waves = 8 batch groups of 16; one block per time segment.
  biquad_wmma_kernel<<<dim3(nseg), dim3(256), 0, stream>>>(clip, out, T);
}